// SphericalHarmonics_60576218743495
// MI455X (gfx1250) — compile-verified
//
#include <hip/hip_runtime.h>
#include <hip/hip_bf16.h>

// ---------------------------------------------------------------------------
// Spherical harmonics (LMAX=4) as GEMM: out[N,50] = mono[N,35] x B^T[35,50]
// B derived at COMPILE TIME by replaying the reference recurrences over a
// constexpr polynomial algebra in monomials x^a y^b z^c, a+b+c <= 4.
// GEMM executed with V_WMMA_F32_16X16X4_F32 (full fp32 precision; kernel is
// HBM-bound at ~212 MB -> ~9 us floor on MI455X, so f32 costs nothing).
//
// Round-1 fix: monomials kept as named scalars (no alloca) so the A-fragment
// half-wave selects stay v_cndmask instead of being promoted to LDS;
// output stores are non-temporal (write-once 200 MB stream).
// ---------------------------------------------------------------------------

#define NMONO 35
#define KPAD  36   // K padded to multiple of 4
#define NPAD  64   // 50 outputs padded to 4 WMMA N-tiles

// ---------------- compile-time math ----------------------------------------
constexpr double csqrt(double x) {
    double g = x > 1.0 ? x : 1.0;
    for (int i = 0; i < 64; ++i) g = 0.5 * (g + x / g);
    return g;
}

// monomial enumeration: degree 0..4; within degree d: a desc, then b desc
constexpr int monoIndex(int a, int b, int c) {
    int d = a + b + c, idx = 0;
    for (int dd = 0; dd < d; ++dd) idx += (dd + 1) * (dd + 2) / 2;
    for (int aa = d; aa > a; --aa) idx += (d - aa + 1);
    idx += (d - a) - b;
    return idx;
}

struct ExpoTab { int a[NMONO]; int b[NMONO]; int c[NMONO]; };
constexpr ExpoTab makeExpoTab() {
    ExpoTab t{};
    int cnt = 0;
    for (int d = 0; d <= 4; ++d)
        for (int a = d; a >= 0; --a)
            for (int b = d - a; b >= 0; --b) {
                t.a[cnt] = a; t.b[cnt] = b; t.c[cnt] = d - a - b;
                ++cnt;
            }
    return t;
}
constexpr ExpoTab ETAB = makeExpoTab();

struct IdxTab { int v[5][5][5]; };
constexpr IdxTab makeIdxTab() {
    IdxTab t{};
    for (int a = 0; a <= 4; ++a)
        for (int b = 0; b + a <= 4; ++b)
            for (int c = 0; a + b + c <= 4; ++c)
                t.v[a][b][c] = monoIndex(a, b, c);
    return t;
}
constexpr IdxTab ITAB = makeIdxTab();

struct Poly { double c[NMONO]; };

constexpr Poly pScale(const Poly& p, double s) {
    Poly r{};
    for (int i = 0; i < NMONO; ++i) r.c[i] = p.c[i] * s;
    return r;
}
constexpr Poly pAdd(const Poly& p, const Poly& q) {
    Poly r{};
    for (int i = 0; i < NMONO; ++i) r.c[i] = p.c[i] + q.c[i];
    return r;
}
constexpr Poly pSub(const Poly& p, const Poly& q) {
    Poly r{};
    for (int i = 0; i < NMONO; ++i) r.c[i] = p.c[i] - q.c[i];
    return r;
}
// multiply by a single monomial x^da y^db z^dc
constexpr Poly pMulMono(const Poly& p, int da, int db, int dc) {
    Poly r{};
    for (int i = 0; i < NMONO; ++i)
        if (p.c[i] != 0.0)
            r.c[ITAB.v[ETAB.a[i] + da][ETAB.b[i] + db][ETAB.c[i] + dc]] = p.c[i];
    return r;
}
constexpr Poly pMul(const Poly& p, const Poly& q) {
    Poly r{};
    for (int i = 0; i < NMONO; ++i) {
        if (p.c[i] == 0.0) continue;
        for (int j = 0; j < NMONO; ++j) {
            if (q.c[j] == 0.0) continue;
            r.c[ITAB.v[ETAB.a[i] + ETAB.a[j]][ETAB.b[i] + ETAB.b[j]][ETAB.c[i] + ETAB.c[j]]]
                += p.c[i] * q.c[j];
        }
    }
    return r;
}

// B table: B[k][j] = coeff of monomial k in output column j (row-major k x 64)
struct BTab { float v[KPAD * NPAD]; };

constexpr BTab buildB() {
    BTab out{};
    constexpr double PI = 3.14159265358979323846;

    double alm[5][5] = {}, blm[5][5] = {};
    for (int l = 1; l <= 4; ++l)
        for (int m = 0; m <= l; ++m) {
            if (m == l) {
                alm[l][m] = -csqrt(1.0 + 0.5 / (double)m);
                blm[l][m] = 0.0;
            } else {
                double lsq = (double)(l * l), ld = 2.0 * l;
                double l1 = 4.0 * lsq - 1.0, l2 = lsq - ld + 1.0;
                alm[l][m] = csqrt(l1 / (lsq - (double)(m * m)));
                blm[l][m] = -csqrt((l2 - (double)(m * m)) / (4.0 * l2 - 1.0));
            }
        }

    Poly plm[5][5] = {};
    plm[0][0].c[0] = csqrt(1.0 / (4.0 * PI));
    plm[1][0].c[monoIndex(0, 0, 1)] = csqrt(3.0 / (4.0 * PI));
    plm[1][1].c[0] = -csqrt(3.0 / (8.0 * PI));
    for (int l = 2; l <= 4; ++l)
        for (int m = 0; m <= l; ++m) {
            if (m == l - 1)
                plm[l][m] = pScale(pMulMono(plm[l - 1][l - 1], 0, 0, 1), csqrt(2.0 * m + 3.0));
            else if (m == l)
                plm[l][m] = pScale(plm[l - 1][l - 1], alm[l][m]);
            else
                plm[l][m] = pScale(pAdd(pMulMono(plm[l - 1][m], 0, 0, 1),
                                        pScale(plm[l - 2][m], blm[l][m])),
                                   alm[l][m]);
        }

    Poly yr[5][5] = {}, yi[5][5] = {};
    for (int l = 0; l <= 4; ++l) yr[l][0] = plm[l][0];  // yi[l][0] stays 0

    Poly phr{}, phi{};
    phr.c[monoIndex(1, 0, 0)] = 1.0;   // x
    phi.c[monoIndex(0, 1, 0)] = 1.0;   // y
    for (int m = 1; m <= 4; ++m) {
        if (m >= 2) {
            Poly pr = phr;
            phr = pSub(pMulMono(phr, 1, 0, 0), pMulMono(phi, 0, 1, 0));
            phi = pAdd(pMulMono(pr, 0, 1, 0), pMulMono(phi, 1, 0, 0));
        }
        for (int l = m; l <= 4; ++l) {
            yr[l][m] = pMul(phr, plm[l][m]);
            yi[l][m] = pMul(phi, plm[l][m]);
        }
    }

    // output columns: t = l^2 + (m + l); col 2t = real, col 2t+1 = imag
    int t = 0;
    for (int l = 0; l <= 4; ++l)
        for (int mm = -l; mm <= l; ++mm) {
            int am = mm < 0 ? -mm : mm;
            double ph = (mm < 0 && (am & 1)) ? -1.0 : 1.0;
            double si = mm < 0 ? -1.0 : 1.0;   // imag gets extra -1 for m<0
            for (int k = 0; k < NMONO; ++k) {
                out.v[k * NPAD + 2 * t]     = (float)(yr[l][am].c[k] * ph);
                out.v[k * NPAD + 2 * t + 1] = (float)(yi[l][am].c[k] * ph * si);
            }
            ++t;
        }
    return out;
}

__constant__ BTab c_B = buildB();

// ---------------- device kernel ---------------------------------------------
typedef __attribute__((ext_vector_type(2))) float v2f;
typedef __attribute__((ext_vector_type(8))) float v8f;

__global__ __launch_bounds__(256)
void ylm_wmma_kernel(const float* __restrict__ rhat, float* __restrict__ out,
                     int npts, int ntiles) {
    const int lane   = threadIdx.x & 31;
    const int j16    = lane & 15;
    const bool hi    = lane >= 16;
    const int  koff  = hi ? 2 : 0;
    const int waveId = blockIdx.x * (blockDim.x >> 5) + (threadIdx.x >> 5);
    const int nwaves = gridDim.x * (blockDim.x >> 5);

    // Hoist constant B fragments: B(4x16) per (k-step, n-tile).
    // Layout mirrors the A 16x4 layout: VGPR0 = rows K0 (lanes 0-15) /
    // K2 (lanes 16-31); VGPR1 = rows K1 / K3.  N = lane%16 within a tile.
    v2f bfrag[9][4];
#pragma unroll
    for (int k = 0; k < 9; ++k)
#pragma unroll
        for (int t = 0; t < 4; ++t) {
            bfrag[k][t].x = c_B.v[(4 * k + koff + 0) * NPAD + t * 16 + j16];
            bfrag[k][t].y = c_B.v[(4 * k + koff + 1) * NPAD + t * 16 + j16];
        }

    for (int tile = waveId; tile < ntiles; tile += nwaves) {
        const int base = tile * 16;
        // lanes L and L+16 both load point (base + L&15): A-matrix row is
        // split K-wise across the two half-waves.
        int p  = base + j16;
        int pc = p < npts ? p : (npts - 1);
        const float x = rhat[3 * pc + 0];
        const float y = rhat[3 * pc + 1];
        const float z = rhat[3 * pc + 2];

        // 35 monomials as NAMED SCALARS (no alloca -> no LDS promotion)
        const float m0 = 1.0f, m1 = x, m2 = y, m3 = z;
        const float m4 = x * x, m5 = x * y, m6 = x * z;
        const float m7 = y * y, m8 = y * z, m9 = z * z;
        const float m10 = m4 * x, m11 = m4 * y, m12 = m4 * z;
        const float m13 = m5 * y, m14 = m5 * z, m15 = m6 * z;
        const float m16 = m7 * y, m17 = m7 * z, m18 = m8 * z, m19 = m9 * z;
        const float m20 = m10 * x, m21 = m10 * y, m22 = m10 * z;
        const float m23 = m11 * y, m24 = m11 * z, m25 = m12 * z;
        const float m26 = m13 * y, m27 = m13 * z, m28 = m14 * z, m29 = m15 * z;
        const float m30 = m16 * y, m31 = m16 * z, m32 = m17 * z, m33 = m18 * z;
        const float m34 = m19 * z, m35 = 0.0f;  // K padding

        // A fragments: lanes 0-15 supply K=4k,4k+1; lanes 16-31 K=4k+2,4k+3.
        // Named-scalar ternaries -> v_cndmask (values stay in VGPRs).
#define MK_A(A, lo0, lo1, hh0, hh1) \
        v2f A; A.x = hi ? (hh0) : (lo0); A.y = hi ? (hh1) : (lo1);
        MK_A(a0, m0,  m1,  m2,  m3)
        MK_A(a1, m4,  m5,  m6,  m7)
        MK_A(a2, m8,  m9,  m10, m11)
        MK_A(a3, m12, m13, m14, m15)
        MK_A(a4, m16, m17, m18, m19)
        MK_A(a5, m20, m21, m22, m23)
        MK_A(a6, m24, m25, m26, m27)
        MK_A(a7, m28, m29, m30, m31)
        MK_A(a8, m32, m33, m34, m35)
#undef MK_A

        v8f acc0 = {}, acc1 = {}, acc2 = {}, acc3 = {};
#define STEP(K, A) \
        acc0 = __builtin_amdgcn_wmma_f32_16x16x4_f32(false, A, false, bfrag[K][0], \
                                                     (short)0, acc0, false, false); \
        acc1 = __builtin_amdgcn_wmma_f32_16x16x4_f32(false, A, false, bfrag[K][1], \
                                                     (short)0, acc1, false, false); \
        acc2 = __builtin_amdgcn_wmma_f32_16x16x4_f32(false, A, false, bfrag[K][2], \
                                                     (short)0, acc2, false, false); \
        acc3 = __builtin_amdgcn_wmma_f32_16x16x4_f32(false, A, false, bfrag[K][3], \
                                                     (short)0, acc3, false, false);
        STEP(0, a0) STEP(1, a1) STEP(2, a2) STEP(3, a3) STEP(4, a4)
        STEP(5, a5) STEP(6, a6) STEP(7, a7) STEP(8, a8)
#undef STEP

        // D layout: VGPR i -> row i (lanes 0-15) / row i+8 (lanes 16-31),
        // column = t*16 + lane%16.  Columns 50..63 are padding.
        // Non-temporal: 200 MB write-once stream, keep it out of L2.
#pragma unroll
        for (int i = 0; i < 8; ++i) {
            const int row = hi ? (i + 8) : i;
            const int pr  = base + row;
            if (pr < npts) {
                float* o = out + (size_t)pr * 50;
                __builtin_nontemporal_store(acc0[i], o + j16);        // cols  0..15
                __builtin_nontemporal_store(acc1[i], o + 16 + j16);   // cols 16..31
                __builtin_nontemporal_store(acc2[i], o + 32 + j16);   // cols 32..47
                if (j16 < 2)
                    __builtin_nontemporal_store(acc3[i], o + 48 + j16); // cols 48..49
            }
        }
    }
}

// ---------------- host launch ------------------------------------------------
extern "C" void kernel_launch(void* const* d_in, const int* in_sizes, int n_in,
                              void* d_out, int out_size, void* d_ws, size_t ws_size,
                              hipStream_t stream) {
    const float* rhat = (const float*)d_in[0];
    float* out = (float*)d_out;
    const int npts   = in_sizes[0] / 3;
    const int ntiles = (npts + 15) / 16;

    dim3 block(256);                 // 8 waves / block (wave32)
    dim3 grid(2048);                 // grid-stride over 62500 tiles
    ylm_wmma_kernel<<<grid, block, 0, stream>>>(rhat, out, npts, ntiles);
}